// SiameseBhatNet_82068235092647
// MI455X (gfx1250) — compile-verified
//
#include <hip/hip_runtime.h>
#include <math.h>

// CDNA5 f32 WMMA operand types: 16x16x4_f32 -> A,B are 2 VGPRs (float2), C/D 8 VGPRs.
typedef float v2f __attribute__((ext_vector_type(2)));
typedef float v8f __attribute__((ext_vector_type(8)));

#define NBATCH 32
#define COUT   256
#define KZ     6
#define MX     22
#define HOUT   17
#define NPOS   (HOUT * HOUT)        // 289
#define SP_X   (MX * MX)            // 484 spatial per sample (x branch)
#define SP_Z   (KZ * KZ)            // 36  spatial per sample (z branch)
#define NELEM_OUT0 (NBATCH * NPOS)  // 9248

// Single-instruction v_sqrt_f32 (TRANS32 -> co-executes with WMMA), vs libm
// sqrtf which expands to ~16 VALU ops of Newton refinement + class checks.
__device__ __forceinline__ float fast_sqrt(float x) {
    return __builtin_amdgcn_sqrtf(x);
}

// ---------------------------------------------------------------------------
// Branch: 1x1 conv (3 -> 256) + bias + ReLU as GEMM via V_WMMA_F32_16X16X4_F32.
//   D[sp, c] = sum_{ci<3} X[n,ci,hw] * Wb[c,ci],  K padded 3->4 with zeros.
// One wave = one 16(spatial) x 16(cout) tile, single WMMA (K fits in one step).
// Total waves = (NB*SPATIAL/16) * (COUT/16); grid sized exactly (8 waves/block).
// ---------------------------------------------------------------------------
template <unsigned SPATIAL>
__global__ void branch_wmma_kernel(const float* __restrict__ X,   // (NB, 3, SPATIAL)
                                   const float* __restrict__ Wb,  // (COUT, 3)
                                   const float* __restrict__ bb,  // (COUT)
                                   float* __restrict__ out)       // (NB, COUT, SPATIAL)
{
    const unsigned wave = (blockIdx.x * blockDim.x + threadIdx.x) >> 5;
    const unsigned lane = threadIdx.x & 31;
    const unsigned c_tile  = wave & 15;   // 16 cout tiles
    const unsigned sp_tile = wave >> 4;

    const unsigned m = lane & 15;         // row within A (spatial)
    const unsigned g = lane >> 4;         // lane half selects K pair {0,1} vs {2,3}

    // ---- A: 16x4 f32.  v0 = K(2g), v1 = K(2g+1); K=3 slot is zero. ----
    const unsigned sp = sp_tile * 16 + m;
    const unsigned n  = sp / SPATIAL;
    const unsigned hw = sp % SPATIAL;
    const unsigned xo = n * 3 * SPATIAL + hw;   // unsigned -> zext -> saddr-form loads
    v2f a;
    a.x = g ? X[xo + 2 * SPATIAL] : X[xo];
    a.y = g ? 0.0f                : X[xo + SPATIAL];

    // ---- B: 4x16 f32 (same K-by-lane-half striping). Column = cout. ----
    const unsigned c = c_tile * 16 + (lane & 15);
    v2f b;
    b.x = g ? Wb[c * 3 + 2] : Wb[c * 3 + 0];
    b.y = g ? 0.0f          : Wb[c * 3 + 1];

    v8f acc = {};
    acc = __builtin_amdgcn_wmma_f32_16x16x4_f32(false, a, false, b,
                                                (short)0, acc, false, false);

    // ---- Epilogue: D layout -> lane holds rows r+8g, column (lane&15). ----
    const float bias = bb[c];
#pragma unroll
    for (unsigned r = 0; r < 8; ++r) {
        const unsigned sp_o = sp_tile * 16 + r + 8 * g;
        const unsigned n_o  = sp_o / SPATIAL;
        const unsigned hw_o = sp_o % SPATIAL;
        float v = acc[r] + bias;
        v = v > 0.0f ? v : 0.0f;     // ReLU
        out[(n_o * COUT + c) * SPATIAL + hw_o] = v;
    }
}

// ---------------------------------------------------------------------------
// Bhattacharyya correlation via WMMA.  Since xf,zf >= 0 (ReLU):
//   coeff[n,pos] = (1/36) * sum_{c,p,q} sqrt(xf[n,c,i+p,j+q]) * (w[c]*sqrt(zf[n,c,p,q]))
// Per-sample GEMV (Kred = 256*36 = 9216) lifted to WMMA 16x16x4_f32 by
// replicating the kernel vector across all 16 B columns; column 0 is the result.
// One wave per (sample, 16-position tile): 19 tiles cover 289 positions.
// Inner 9-chunk loop fully unrolled so (p,q) offsets const-fold (chunk starts
// are even -> each f32 pair stays inside one patch row).
// Bases are wave-uniform (SGPR pair); per-lane indices are UNSIGNED 32-bit so
// the zext(base64 + u32) pattern selects GVS saddr-form loads
// (global_load_b64 v, v_off, s[base]) with no 64-bit vector address math.
// Per chunk: 2x global_load_b64 + 4x v_sqrt_f32 (TRANS, overlaps WMMA)
//            + muls + 1x v_wmma_f32_16x16x4_f32.
// ---------------------------------------------------------------------------
__global__ void corr_wmma_kernel(const float* __restrict__ xf,  // (NB, COUT, 22, 22)
                                 const float* __restrict__ zf,  // (NB, COUT, 6, 6)
                                 const float* __restrict__ w,   // (COUT)
                                 float* __restrict__ out0)      // (NB, 289) unnormalized
{
    const unsigned lane = threadIdx.x;   // blockDim = 32, one wave
    const unsigned tile = blockIdx.x;    // 0..18
    const unsigned n    = blockIdx.y;    // 0..31

    const unsigned m = lane & 15;
    const unsigned g = lane >> 4;

    unsigned pos = tile * 16 + m;
    if (pos > NPOS - 1) pos = NPOS - 1;  // clamp (no divergence; dup rows, not stored)
    const unsigned i = pos / HOUT;
    const unsigned j = pos % HOUT;

    const float* __restrict__ xf_n = xf + n * COUT * SP_X;  // wave-uniform base (SGPR)
    const float* __restrict__ zf_n = zf + n * COUT * SP_Z;  // wave-uniform base (SGPR)
    const unsigned xoff0 = i * MX + j;                      // per-lane u32 offset

    v8f acc = {};
    for (unsigned c = 0; c < COUT; ++c) {
        const unsigned xbase = c * SP_X + xoff0;  // u32 voffset, +484 per iter
        const unsigned zbase = c * SP_Z;          // u32 voffset, +36  per iter
        const float wc  = w[c];
#pragma unroll
        for (unsigned r = 0; r < 9; ++r) {        // 9 chunks of K=4 cover 36 patch slots
            const unsigned kA = r * 4;            // lane half g=0 takes (kA, kA+1)
            const unsigned kB = r * 4 + 2;        // lane half g=1 takes (kB, kB+1)
            const unsigned offA = (kA / KZ) * MX + (kA % KZ);
            const unsigned offB = (kB / KZ) * MX + (kB % KZ);
            const unsigned off = g ? offB : offA; // compile-time consts -> v_cndmask
            const unsigned kk  = g ? kB  : kA;
            v2f a, b;
            a.x = fast_sqrt(xf_n[xbase + off]);
            a.y = fast_sqrt(xf_n[xbase + off + 1]);  // same patch row (kk even, KZ even)
            b.x = wc * fast_sqrt(zf_n[zbase + kk]);
            b.y = wc * fast_sqrt(zf_n[zbase + kk + 1]);
            acc = __builtin_amdgcn_wmma_f32_16x16x4_f32(false, a, false, b,
                                                        (short)0, acc, false, false);
        }
    }

    // Column 0 of D lives in lanes 0 (rows 0..7) and 16 (rows 8..15).
    if ((lane & 15) == 0) {
#pragma unroll
        for (unsigned r = 0; r < 8; ++r) {
            const unsigned posw = tile * 16 + r + 8 * g;
            if (posw < NPOS)
                out0[n * NPOS + posw] = acc[r] * (1.0f / 36.0f);
        }
    }
}

// ---------------------------------------------------------------------------
// Training-mode BatchNorm2d over the single channel: batch stats across all
// 9248 scores, biased variance, then in-place normalize. One block.
// ---------------------------------------------------------------------------
__global__ void bn_kernel(float* __restrict__ out0,
                          const float* __restrict__ gamma,
                          const float* __restrict__ beta)
{
    __shared__ float s1[256];
    __shared__ float s2[256];
    const unsigned t = threadIdx.x;

    float a1 = 0.0f, a2 = 0.0f;
    for (unsigned idx = t; idx < NELEM_OUT0; idx += 256) {
        const float v = out0[idx];
        a1 += v;
        a2 += v * v;
    }
    s1[t] = a1; s2[t] = a2;
    __syncthreads();
    for (unsigned s = 128; s > 0; s >>= 1) {
        if (t < s) { s1[t] += s1[t + s]; s2[t] += s2[t + s]; }
        __syncthreads();
    }
    const float inv_n = 1.0f / (float)NELEM_OUT0;
    const float mu  = s1[0] * inv_n;
    const float var = s2[0] * inv_n - mu * mu;        // biased variance
    const float sc  = gamma[0] / fast_sqrt(var + 1e-5f);
    const float sh  = beta[0] - mu * sc;
    for (unsigned idx = t; idx < NELEM_OUT0; idx += 256)
        out0[idx] = out0[idx] * sc + sh;
}

// ---------------------------------------------------------------------------
// d_in order: z, x, Wb, bb, w, gamma, beta (all f32).
// d_out: [out (32*289)] [xf (32*256*484)] [zf (32*256*36)] concatenated.
// ---------------------------------------------------------------------------
extern "C" void kernel_launch(void* const* d_in, const int* in_sizes, int n_in,
                              void* d_out, int out_size, void* d_ws, size_t ws_size,
                              hipStream_t stream) {
    const float* z     = (const float*)d_in[0];
    const float* x     = (const float*)d_in[1];
    const float* Wb    = (const float*)d_in[2];
    const float* bb    = (const float*)d_in[3];
    const float* w     = (const float*)d_in[4];
    const float* gamma = (const float*)d_in[5];
    const float* beta  = (const float*)d_in[6];

    float* out0 = (float*)d_out;                         // (32,1,17,17)
    float* xfp  = out0 + NELEM_OUT0;                     // (32,256,22,22)
    float* zfp  = xfp + (size_t)NBATCH * COUT * SP_X;    // (32,256,6,6)

    // x branch: 15488 spatial rows -> 968 sp-tiles * 16 c-tiles = 15488 waves
    branch_wmma_kernel<SP_X><<<dim3(15488 * 16 / 8 / 16), 256, 0, stream>>>(x, Wb, bb, xfp);
    // z branch: 1152 spatial rows -> 72 sp-tiles * 16 c-tiles = 1152 waves
    branch_wmma_kernel<SP_Z><<<dim3(1152 * 16 / 8 / 16), 256, 0, stream>>>(z, Wb, bb, zfp);
    // correlation: one wave per (position-tile, sample)
    corr_wmma_kernel<<<dim3(19, NBATCH), 32, 0, stream>>>(xfp, zfp, w, out0);
    // batchnorm (single block, in-place)
    bn_kernel<<<1, 256, 0, stream>>>(out0, gamma, beta);
}